// GCN_87514253623372
// MI455X (gfx1250) — compile-verified
//
#include <hip/hip_runtime.h>
#include <hip/hip_bf16.h>

#define NN 50000
#define NE 1200000
#define HD 64
// pair-interleaved LDS layout for W: sWp[(k>>1)*PSTR + 2*n + (k&1)]
// PSTR=160 floats => g=1 half-wave offset 160 ≡ 32 (mod 64 banks): conflict-free b64 reads
#define PSTR 160

typedef float v2f __attribute__((ext_vector_type(2)));
typedef float v8f __attribute__((ext_vector_type(8)));

// ---------------- utility kernels ----------------

__global__ void fill_f32(float* __restrict__ p, float v, int n) {
  int stride = gridDim.x * blockDim.x;
  for (int i = blockIdx.x * blockDim.x + threadIdx.x; i < n; i += stride) p[i] = v;
}

__global__ void degree_kernel(const long long* __restrict__ ei, float* __restrict__ deg, int nE) {
  int stride = gridDim.x * blockDim.x;
  for (int i = blockIdx.x * blockDim.x + threadIdx.x; i < nE; i += stride)
    atomicAdd(&deg[(int)ei[i]], 1.0f);
}

__global__ void rsqrt_kernel(float* __restrict__ d, int n) {
  int stride = gridDim.x * blockDim.x;
  for (int i = blockIdx.x * blockDim.x + threadIdx.x; i < n; i += stride)
    d[i] = rsqrtf(d[i]);
}

// norm[e] = dis[u]*dis[v]; also narrow the int64 edge index to int32 once.
__global__ void norm_kernel(const long long* __restrict__ ei, const float* __restrict__ dis,
                            float* __restrict__ nrm, int* __restrict__ u32,
                            int* __restrict__ v32, int nE) {
  int stride = gridDim.x * blockDim.x;
  for (int e = blockIdx.x * blockDim.x + threadIdx.x; e < nE; e += stride) {
    int u = (int)ei[e];
    int v = (int)ei[nE + e];
    u32[e] = u;
    v32[e] = v;
    nrm[e] = dis[u] * dis[v];
  }
}

// one wave per edge: lane handles 2 of the 64 features
__global__ void scatter_kernel(const int* __restrict__ u32, const int* __restrict__ v32,
                               const float* __restrict__ nrm, const float* __restrict__ H,
                               float* __restrict__ agg, int nE) {
  int lane = threadIdx.x & 31;
  int warp = (blockIdx.x * blockDim.x + threadIdx.x) >> 5;
  int nWarp = (gridDim.x * blockDim.x) >> 5;
  for (int e = warp; e < nE; e += nWarp) {
    float s = nrm[e];
    const float* src = H + (size_t)u32[e] * HD;
    float* dst = agg + (size_t)v32[e] * HD;
    float2 h = *(const float2*)(src + lane * 2);
    atomicAdd(dst + lane * 2 + 0, s * h.x);
    atomicAdd(dst + lane * 2 + 1, s * h.y);
  }
}

// ---------------- WMMA fp32 GEMM: Out[M x 64] = act((A (+A2)) @ W + bias) ----------------
// per wave: one 16x64 output tile; 16 K-steps of V_WMMA_F32_16X16X4_F32 x 4 N-tiles
template <bool HAS_A2, bool DO_RELU>
__global__ __launch_bounds__(128) void gemm_wmma_64(const float* __restrict__ A,
                                                    const float* __restrict__ A2,
                                                    const float* __restrict__ W,
                                                    const float* __restrict__ bias,
                                                    float* __restrict__ Out, int M) {
  __shared__ float sWp[32 * PSTR];  // 20 KB, pair-interleaved
  __shared__ float sB[HD];
  for (int i = threadIdx.x; i < HD * HD; i += blockDim.x) {
    int k = i >> 6, n = i & 63;
    sWp[(k >> 1) * PSTR + 2 * n + (k & 1)] = W[i];
  }
  if (threadIdx.x < HD) sB[threadIdx.x] = bias[threadIdx.x];
  __syncthreads();

  const int lane = threadIdx.x & 31;
  const int wave = threadIdx.x >> 5;
  const int tile = blockIdx.x * 4 + wave;  // wave-uniform
  if (tile * 16 >= M) return;              // uniform exit, EXEC stays full
  const int row0 = tile * 16;
  const int m = lane & 15;  // A row within tile / B-and-D column
  const int g = lane >> 4;  // lane-half selector

  const float* arow = A + (size_t)(row0 + m) * HD;
  const float* arow2 = HAS_A2 ? (A2 + (size_t)(row0 + m) * HD) : nullptr;
  // this lane's B base for k-pair (2s+g): sWp + (2s+g)*PSTR + 2m
  const float* wb = sWp + (size_t)g * PSTR + 2 * m;

  v8f c0 = {}, c1 = {}, c2 = {}, c3 = {};
#pragma unroll
  for (int s = 0; s < 16; ++s) {
    const int k0 = 4 * s + 2 * g;  // this lane's K pair
    float2 av = *(const float2*)(arow + k0);
    if (HAS_A2) {
      float2 av2 = *(const float2*)(arow2 + k0);
      av.x += av2.x;
      av.y += av2.y;
    }
    v2f a = {av.x, av.y};
    const float* wk = wb + 2 * s * PSTR;
    v2f b0 = *(const v2f*)(wk + 0);
    c0 = __builtin_amdgcn_wmma_f32_16x16x4_f32(false, a, false, b0, (short)0, c0, false, false);
    v2f b1 = *(const v2f*)(wk + 32);
    c1 = __builtin_amdgcn_wmma_f32_16x16x4_f32(false, a, false, b1, (short)0, c1, false, false);
    v2f b2 = *(const v2f*)(wk + 64);
    c2 = __builtin_amdgcn_wmma_f32_16x16x4_f32(false, a, false, b2, (short)0, c2, false, false);
    v2f b3 = *(const v2f*)(wk + 96);
    c3 = __builtin_amdgcn_wmma_f32_16x16x4_f32(false, a, false, b3, (short)0, c3, false, false);
  }

  // D layout: vgpr j, lane-half g -> row g*8+j, col lane&15 (per 16-wide N tile)
  float* obase = Out + (size_t)row0 * HD;
  const int rb = g * 8;
#pragma unroll
  for (int j = 0; j < 8; ++j) {
    float* orow = obase + (size_t)(rb + j) * HD;
    float v0 = c0[j] + sB[m + 0];
    float v1 = c1[j] + sB[m + 16];
    float v2 = c2[j] + sB[m + 32];
    float v3 = c3[j] + sB[m + 48];
    if (DO_RELU) {
      v0 = fmaxf(v0, 0.f);
      v1 = fmaxf(v1, 0.f);
      v2 = fmaxf(v2, 0.f);
      v3 = fmaxf(v3, 0.f);
    }
    orow[m + 0] = v0;
    orow[m + 16] = v1;
    orow[m + 32] = v2;
    orow[m + 48] = v3;
  }
}

// ---------------- readout ----------------

__global__ void colsum_kernel(const float* __restrict__ Hf, float* __restrict__ colsum, int nRows) {
  int col = threadIdx.x & 63;
  int sub = threadIdx.x >> 6;
  int spb = blockDim.x >> 6;
  float acc = 0.f;
  for (int r = blockIdx.x * spb + sub; r < nRows; r += gridDim.x * spb)
    acc += Hf[(size_t)r * HD + col];
  atomicAdd(&colsum[col], acc);
}

__global__ void final_kernel(const float* __restrict__ colsum, const float* __restrict__ out_w,
                             const float* __restrict__ out_b, float* __restrict__ out) {
  __shared__ float red[64];
  int t = threadIdx.x;
  red[t] = colsum[t] * (1.0f / (float)NN) * out_w[t];
  __syncthreads();
  for (int s = 32; s > 0; s >>= 1) {
    if (t < s) red[t] += red[t + s];
    __syncthreads();
  }
  if (t == 0) out[0] = red[0] + out_b[0];
}

// ---------------- launcher ----------------

extern "C" void kernel_launch(void* const* d_in, const int* in_sizes, int n_in,
                              void* d_out, int out_size, void* d_ws, size_t ws_size,
                              hipStream_t stream) {
  (void)in_sizes; (void)n_in; (void)out_size; (void)ws_size;

  const float* H = (const float*)d_in[0];
  const long long* ei = (const long long*)d_in[1];  // int64 [2, NE]
  // d_in[2] = E (unused by reference)
  const float* enc_w[3] = {(const float*)d_in[3], (const float*)d_in[7], (const float*)d_in[11]};
  const float* enc_b[3] = {(const float*)d_in[4], (const float*)d_in[8], (const float*)d_in[12]};
  const float* upd_w[3] = {(const float*)d_in[5], (const float*)d_in[9], (const float*)d_in[13]};
  const float* upd_b[3] = {(const float*)d_in[6], (const float*)d_in[10], (const float*)d_in[14]};
  const float* out_w = (const float*)d_in[15];
  const float* out_b = (const float*)d_in[16];

  char* ws = (char*)d_ws;
  size_t off = 0;
  auto carve = [&](size_t bytes) -> void* {
    void* p = ws + off;
    off = (off + bytes + 255) & ~(size_t)255;
    return p;
  };
  float* deg = (float*)carve((size_t)NN * 4);  // becomes dis after rsqrt
  float* nrm = (float*)carve((size_t)NE * 4);
  int* u32 = (int*)carve((size_t)NE * 4);
  int* v32 = (int*)carve((size_t)NE * 4);
  float* Ha = (float*)carve((size_t)NN * HD * 4);  // encoder output
  float* Hb = (float*)carve((size_t)NN * HD * 4);  // layer output
  float* agg = (float*)carve((size_t)NN * HD * 4);
  float* colsum = (float*)carve(64 * 4);

  // normalization: deg = 1 + segment_sum(ones, u); dis = deg^-0.5; norm = dis[u]*dis[v]
  fill_f32<<<196, 256, 0, stream>>>(deg, 1.0f, NN);
  degree_kernel<<<4688, 256, 0, stream>>>(ei, deg, NE);
  rsqrt_kernel<<<196, 256, 0, stream>>>(deg, NN);
  norm_kernel<<<4688, 256, 0, stream>>>(ei, deg, nrm, u32, v32, NE);

  const int gemmBlocks = (NN / 16 + 3) / 4;  // 3125 tiles -> 782 blocks of 4 waves
  const int scatBlocks = NE / 8;             // 8 warps per 256-thread block

  const float* Hcur = H;
  for (int l = 0; l < 3; ++l) {
    gemm_wmma_64<false, false>
        <<<gemmBlocks, 128, 0, stream>>>(Hcur, nullptr, enc_w[l], enc_b[l], Ha, NN);
    fill_f32<<<12500, 256, 0, stream>>>(agg, 0.0f, NN * HD);
    scatter_kernel<<<scatBlocks, 256, 0, stream>>>(u32, v32, nrm, Ha, agg, NE);
    gemm_wmma_64<true, true>
        <<<gemmBlocks, 128, 0, stream>>>(agg, Ha, upd_w[l], upd_b[l], Hb, NN);
    Hcur = Hb;
  }

  fill_f32<<<1, 64, 0, stream>>>(colsum, 0.0f, 64);
  colsum_kernel<<<128, 256, 0, stream>>>(Hb, colsum, NN);
  final_kernel<<<1, 64, 0, stream>>>(colsum, out_w, out_b, (float*)d_out);
}